// reconstruction_loss_31568009626038
// MI455X (gfx1250) — compile-verified
//
#include <hip/hip_runtime.h>
#include <hip/hip_bf16.h>

// Charbonnier mean over MSFA-gathered pixels.
// X, Y: [B=8, C=16, H=512, W=512] f32. Output: scalar f32.
// Bandwidth-bound strided gather (~64 MB line traffic -> ~2.7us @ 23.3 TB/s).
// Intra-wave reductions use V_WMMA_F32_16X16X4_F32 (exact f32 row-sum trick).
// sqrt via raw v_sqrt_f32: argument is always a positive normal (>= 1e-6),
// so the IEEE special-case expansion of sqrtf() is unnecessary VALU work.

typedef __attribute__((ext_vector_type(2))) float v2f;
typedef __attribute__((ext_vector_type(8))) float v8f;

#define N_TOTAL 2097152u   // 8*512*512
#define HW_SH   18         // H*W = 2^18
#define BLOCKS  1024
#define TPB     256

// Exact f32 sum of all 32 lane values, result broadcast to every lane.
// A VGPR0 = per-lane value (covers 32 distinct A[m,k] slots), A VGPR1 = 0,
// B = all ones (layout-independent). D[m,n] = rowsum_m. Each lane holds 8
// distinct M rows of one column; summing its D regs gives half-sum, xor-16
// shuffle-add completes the total.
__device__ __forceinline__ float wave_sum_wmma(float v) {
    v2f a; a.x = v;    a.y = 0.0f;
    v2f b; b.x = 1.0f; b.y = 1.0f;
    v8f c = {};
    v8f d = __builtin_amdgcn_wmma_f32_16x16x4_f32(
        /*neg_a=*/false, a, /*neg_b=*/false, b,
        /*c_mod=*/(short)0, c, /*reuse_a=*/false, /*reuse_b=*/false);
    float s = ((d[0] + d[1]) + (d[2] + d[3])) + ((d[4] + d[5]) + (d[6] + d[7]));
    s += __shfl_xor(s, 16, 32);
    return s;
}

__global__ void __launch_bounds__(TPB)
msfa_charb_partial(const float* __restrict__ X, const float* __restrict__ Y,
                   float* __restrict__ ws) {
    const unsigned tid  = threadIdx.x;
    const unsigned base = blockIdx.x * TPB + tid;      // rem = h*512 + w in [0, 2^18)
    const unsigned h    = base >> 9;
    const unsigned w    = base & 511u;
    const unsigned ch   = ((h & 3u) << 2) | (w & 3u);  // mosaic channel

    float acc = 0.0f;
#pragma unroll
    for (unsigned b = 0; b < 8u; ++b) {
        // flat index of x[b, ch, h, w] = ((b*16 + ch) << 18) + rem
        size_t off = ((size_t)((b << 4) + ch) << HW_SH) + base;
        float dx = X[off] - Y[off];
        acc += __builtin_amdgcn_sqrtf(__builtin_fmaf(dx, dx, 1e-6f));
    }

    // wave-level exact reduction via WMMA (EXEC is all-ones: no divergence here)
    float wsum = wave_sum_wmma(acc);

    __shared__ float lds[TPB / 32];
    if ((tid & 31u) == 0u) lds[tid >> 5] = wsum;
    __syncthreads();

    if (tid < 32u) {                      // wave-uniform branch: wave 0, full EXEC
        float v = (tid < (TPB / 32)) ? lds[tid] : 0.0f;
        float bsum = wave_sum_wmma(v);
        if (tid == 0u) ws[blockIdx.x] = bsum;
    }
}

__global__ void __launch_bounds__(TPB)
msfa_charb_final(const float* __restrict__ ws, float* __restrict__ out) {
    const unsigned tid = threadIdx.x;
    float v = (ws[tid] + ws[tid + 256]) + (ws[tid + 512] + ws[tid + 768]);

    float wsum = wave_sum_wmma(v);

    __shared__ float lds[TPB / 32];
    if ((tid & 31u) == 0u) lds[tid >> 5] = wsum;
    __syncthreads();

    if (tid < 32u) {
        float vv = (tid < (TPB / 32)) ? lds[tid] : 0.0f;
        float tot = wave_sum_wmma(vv);
        if (tid == 0u) out[0] = tot * (1.0f / (float)N_TOTAL);
    }
}

extern "C" void kernel_launch(void* const* d_in, const int* in_sizes, int n_in,
                              void* d_out, int out_size, void* d_ws, size_t ws_size,
                              hipStream_t stream) {
    const float* X = (const float*)d_in[0];
    const float* Y = (const float*)d_in[1];
    float* ws  = (float*)d_ws;   // needs 1024 floats (4 KB)
    float* out = (float*)d_out;

    msfa_charb_partial<<<BLOCKS, TPB, 0, stream>>>(X, Y, ws);
    msfa_charb_final<<<1, TPB, 0, stream>>>(ws, out);
}